// OnePoleDecoder_82695300317749
// MI455X (gfx1250) — compile-verified
//
#include <hip/hip_runtime.h>
#include <hip/hip_bf16.h>

typedef __attribute__((ext_vector_type(2))) float v2f;
typedef __attribute__((ext_vector_type(8))) float v8f;
typedef int v4i __attribute__((vector_size(16)));

#define T_TOTAL   131072
#define IN_DIM    64
#define HID       512
#define T_CHUNK   1024
#define HALO      256                      // alpha^256 = e^-32 ~ 1.3e-14 << f32 eps
#define NROWS     (T_CHUNK + HALO)         // 1280
#define NBLK      (NROWS / 16)             // 80
#define NWAVES    8
#define BLOCK_THREADS (NWAVES * 32)        // 256
#define ALPHA_F   0.88249690258459546f     // exp(-1/8)

#if defined(__has_builtin)
#if __has_builtin(__builtin_amdgcn_global_load_async_to_lds_b128) && \
    __has_builtin(__builtin_amdgcn_s_wait_asynccnt)
#define USE_ASYNC_STAGE 1
#endif
#endif
#ifndef USE_ASYNC_STAGE
#define USE_ASYNC_STAGE 0
#endif

// Native CDNA5 V_TANH_F32 when available; branchless exp2-based fallback otherwise.
__device__ __forceinline__ float fast_tanh(float x) {
#if defined(__has_builtin) && __has_builtin(__builtin_amdgcn_tanhf)
    return __builtin_amdgcn_tanhf(x);
#else
    float xc = fminf(fmaxf(x, -9.0f), 9.0f);
    float e  = __builtin_amdgcn_exp2f(xc * 2.885390081777927f);  // exp(2x)
    return (e - 1.0f) / (e + 1.0f);
#endif
}

__global__ __launch_bounds__(BLOCK_THREADS)
void OnePoleDecoder_kernel(const float* __restrict__ beliefs,
                           const float* __restrict__ W_in,
                           const float* __restrict__ b_in,
                           const float* __restrict__ w_theta,
                           const float* __restrict__ b_theta,
                           const float* __restrict__ w_z,
                           const float* __restrict__ b_z,
                           float* __restrict__ out_theta,
                           float* __restrict__ out_z)
{
    __shared__ float Ws[IN_DIM * HID];     // 128 KB, row-major copy of W_in
    __shared__ float bil[HID];
    __shared__ float wtl[HID];
    __shared__ float wzl[HID];
    __shared__ float ubuf[NROWS];
    __shared__ float vbuf[NROWS];

    const int tid  = threadIdx.x;
    const int wid  = tid >> 5;
    const int lane = tid & 31;
    const int hi   = lane >> 4;    // half-wave select
    const int ln   = lane & 15;
    const int chunkStart = (int)blockIdx.x * T_CHUNK;

    // ---- stage W_in into LDS ----
#if USE_ASYNC_STAGE
    // CDNA5 async copy: GLOBAL_LOAD_ASYNC_TO_LDS_B128, tracked by ASYNCcnt.
    for (int c = tid; c < (IN_DIM * HID) / 4; c += BLOCK_THREADS) {
        __builtin_amdgcn_global_load_async_to_lds_b128(
            (v4i*)(W_in + 4 * c), (v4i*)(Ws + 4 * c), 0, 0);
    }
    __builtin_amdgcn_s_wait_asynccnt(0);
#else
    for (int c = tid; c < IN_DIM * HID; c += BLOCK_THREADS)
        Ws[c] = W_in[c];
#endif
    for (int n = tid; n < HID; n += BLOCK_THREADS) {
        bil[n] = b_in[n];
        wtl[n] = w_theta[n];
        wzl[n] = w_z[n];
    }
    __syncthreads();

    // ---- phase 1: per 16-row tile, X = tanh(A@W + b); u = X.w_theta, v = X.w_z ----
    for (int rb = wid; rb < NBLK; rb += NWAVES) {      // uniform per wave: EXEC all-ones
        const int r0 = chunkStart - HALO + rb * 16;    // may be < 0 for block 0 halo
        int row = r0 + ln;
        int rc  = row < 0 ? 0 : row;                   // clamp (halo result overwritten w/ 0)
        const float* arow = beliefs + (size_t)rc * IN_DIM;

        // A fragments for the whole K=64: 16 k-steps of 4, held in registers
        v2f afrag[16];
#pragma unroll
        for (int kk = 0; kk < 16; ++kk)
            afrag[kk] = *(const v2f*)(arow + kk * 4 + 2 * hi);

        float uacc[8], vacc[8];
#pragma unroll
        for (int i = 0; i < 8; ++i) { uacc[i] = 0.f; vacc[i] = 0.f; }

        for (int ct = 0; ct < HID / 16; ++ct) {
            const int n0 = ct * 16;
            const float bi = bil[n0 + ln];
            const float wt = wtl[n0 + ln];
            const float wv = wzl[n0 + ln];
            v8f c = { bi, bi, bi, bi, bi, bi, bi, bi };   // C(m,n) starts at b_in[n]
#pragma unroll
            for (int kk = 0; kk < 16; ++kk) {
                const int k = kk * 4 + 2 * hi;            // B K-rows k, k+1 for this half
                v2f b;
                b.x = Ws[k * HID + n0 + ln];
                b.y = Ws[(k + 1) * HID + n0 + ln];
                c = __builtin_amdgcn_wmma_f32_16x16x4_f32(
                        false, afrag[kk], false, b, (short)0, c, false, false);
            }
            // tanh + fused projection onto the two heads
#pragma unroll
            for (int i = 0; i < 8; ++i) {
                float x = fast_tanh(c[i]);
                uacc[i] += x * wt;
                vacc[i] += x * wv;
            }
        }
        // reduce over the 16 lanes of each half-wave (xor<=8 stays within halves)
#pragma unroll
        for (int i = 0; i < 8; ++i) {
#pragma unroll
            for (int m = 1; m <= 8; m <<= 1) {
                uacc[i] += __shfl_xor(uacc[i], m, 32);
                vacc[i] += __shfl_xor(vacc[i], m, 32);
            }
        }
        if (ln == 0) {   // lane 0 -> rows 0..7, lane 16 -> rows 8..15
#pragma unroll
            for (int i = 0; i < 8; ++i) {
                int m = hi * 8 + i;
                int t = r0 + m;
                ubuf[rb * 16 + m] = (t < 0) ? 0.f : uacc[i];
                vbuf[rb * 16 + m] = (t < 0) ? 0.f : vacc[i];
            }
        }
    }
    __syncthreads();

    // ---- phase 2: weighted Kogge-Stone scan (EMA commutes with the linear heads) ----
    if (wid == 0) {
        const float a1    = ALPHA_F;
        const float oma   = 1.0f - a1;
        // alpha^(lane+1) = exp2(-0.125*log2e*(lane+1))
        const float alane = __builtin_amdgcn_exp2f(-0.180336880111160f * (float)(lane + 1));
        const float bth   = b_theta[0];
        const float bzz   = b_z[0];
        float cu = 0.f, cv = 0.f;
        for (int blk = 0; blk < NROWS / 32; ++blk) {
            const int i = blk * 32 + lane;
            float su = oma * ubuf[i];
            float sv = oma * vbuf[i];
            float w = a1;                                // alpha^d, d = 1,2,4,8,16
#pragma unroll
            for (int d = 1; d < 32; d <<= 1) {
                float tu = __shfl_up(su, d, 32);
                float tv = __shfl_up(sv, d, 32);
                if (lane >= d) { su += w * tu; sv += w * tv; }
                w = w * w;
            }
            float hu = su + alane * cu;
            float hv = sv + alane * cv;
            int t = chunkStart - HALO + i;
            if (t >= chunkStart) {
                out_theta[t] = hu + bth;
                out_z[t]     = hv + bzz;
            }
            cu = __shfl(hu, 31, 32);
            cv = __shfl(hv, 31, 32);
        }
    }
}

extern "C" void kernel_launch(void* const* d_in, const int* in_sizes, int n_in,
                              void* d_out, int out_size, void* d_ws, size_t ws_size,
                              hipStream_t stream) {
    const float* beliefs = (const float*)d_in[0];
    const float* W_in    = (const float*)d_in[1];
    const float* b_in    = (const float*)d_in[2];
    const float* w_theta = (const float*)d_in[3];
    const float* b_theta = (const float*)d_in[4];
    const float* w_z     = (const float*)d_in[5];
    const float* b_z     = (const float*)d_in[6];
    float* out_theta = (float*)d_out;
    float* out_z     = out_theta + T_TOTAL;

    dim3 grid(T_TOTAL / T_CHUNK);     // 128 workgroups
    dim3 block(BLOCK_THREADS);        // 8 wave32
    OnePoleDecoder_kernel<<<grid, block, 0, stream>>>(
        beliefs, W_in, b_in, w_theta, b_theta, w_z, b_z, out_theta, out_z);
}